// gMLPBlock_11819749999046
// MI455X (gfx1250) — compile-verified
//
#include <hip/hip_runtime.h>
#include <math.h>

typedef __attribute__((ext_vector_type(2))) float v2f;
typedef __attribute__((ext_vector_type(8))) float v8f;

#define LN_EPS 1e-5f

// Branchless erf (njuffa-style two-range minimax, ~1 ulp), both ranges computed,
// selected with one cndmask -- no EXEC divergence, no s_cbranch inside the hot loop.
__device__ __forceinline__ float erf_branchless(float a) {
    const float t = fabsf(a);
    const float s = a * a;
    // |a| <= 0.921875 : polynomial in a^2, odd function via final fma with a
    float p = -5.96761703e-4f;
    p = fmaf(p, s, 4.99119423e-3f);
    p = fmaf(p, s, -2.67681349e-2f);
    p = fmaf(p, s, 1.12819925e-1f);
    p = fmaf(p, s, -3.76125336e-1f);
    p = fmaf(p, s, 1.28379166e-1f);
    p = fmaf(p, a, a);
    // |a| > 0.921875 : 1 - exp(poly(t)), sign restored
    float q = fmaf(-1.72853470e-5f, t, 3.83197126e-4f);
    float u = fmaf(-3.88396438e-3f, t, 2.42546219e-2f);
    q = fmaf(q, s, u);
    q = fmaf(q, t, -1.06777877e-1f);
    q = fmaf(q, t, -6.34846687e-1f);
    q = fmaf(q, t, -1.28717512e-1f);
    q = fmaf(q, t, -t);
    q = 1.0f - __expf(q);
    q = copysignf(q, a);
    return (t > 0.921875f) ? q : p;
}

__device__ __forceinline__ float gelu_exact(float x) {
    // exact GELU: 0.5*x*(1+erf(x/sqrt(2)))
    return 0.5f * x * (1.0f + erf_branchless(x * 0.70710678118654752440f));
}

__global__ __launch_bounds__(256) void gmlp_fused_kernel(
    const float* __restrict__ x,
    const float* __restrict__ ln1_g, const float* __restrict__ ln1_b,
    const float* __restrict__ W1,    const float* __restrict__ b1,
    const float* __restrict__ sgu_g, const float* __restrict__ sgu_b,
    const float* __restrict__ conv_w,const float* __restrict__ conv_b,
    const float* __restrict__ W2,    const float* __restrict__ b2,
    float* __restrict__ out, int B)
{
    // 2KB transpose slab per wave (8 waves/block -> 16KB of the 320KB WGP LDS)
    __shared__ float lds[8 * 32 * 16];

    const unsigned tid  = threadIdx.x;
    const unsigned lane = tid & 31u;           // position p within sample
    const unsigned wid  = tid >> 5;            // wave within block
    const unsigned l15  = lane & 15u;
    const bool     hi   = lane >= 16u;
    float* tile = &lds[wid * 512];

    // per-position LN affine parameters (LN is over the 32-position axis)
    const float g1  = ln1_g[lane], be1 = ln1_b[lane];
    const float g2  = sgu_g[lane], be2 = sgu_b[lane];

    // ---- stationary WMMA B operands (4x16 f32 layout:
    //      VGPR0 = K0 (lanes 0-15) / K2 (lanes 16-31); VGPR1 = K1 / K3).
    //      B[k][n] = W[n][k]; columns n>=8 zero-padded. ----
    v2f Bw1lo, Bw1hi, Bw2;
    {
        float a0=0.f,a1=0.f,a2=0.f,a3=0.f,a4=0.f,a5=0.f;
        if (l15 < 8u) {
            const float* r1 = W1 + l15 * 8;        // W1 is (8,8) row-major
            a0 = r1[hi ? 2 : 0]; a1 = r1[hi ? 3 : 1];
            a2 = r1[hi ? 6 : 4]; a3 = r1[hi ? 7 : 5];
            const float* r2 = W2 + l15 * 4;        // W2 is (8,4) row-major
            a4 = r2[hi ? 2 : 0]; a5 = r2[hi ? 3 : 1];
        }
        Bw1lo = (v2f){a0, a1};
        Bw1hi = (v2f){a2, a3};
        Bw2   = (v2f){a4, a5};
    }
    // biases broadcast into the C matrix (C layout: lane -> column N = lane&15)
    const float cbias1 = (l15 < 8u) ? b1[l15] : 0.f;
    const float cbias2 = (l15 < 8u) ? b2[l15] : 0.f;

    // conv weights (uniform; compiler keeps them scalar / hoisted)
    float cw[4][4][5];
    float cb[4];
    #pragma unroll
    for (int o = 0; o < 4; ++o) {
        cb[o] = conv_b[o];
        #pragma unroll
        for (int i = 0; i < 4; ++i)
            #pragma unroll
            for (int t = 0; t < 5; ++t)
                cw[o][i][t] = conv_w[(o * 4 + i) * 5 + t];
    }

    const int wavesTotal = (int)(gridDim.x * (blockDim.x >> 5));
    const int wave       = (int)(blockIdx.x * (blockDim.x >> 5) + wid);
    const int src0 = (int)l15;          // A-source lane, M-tile 0 (pos 0..15)
    const int src1 = (int)(l15 | 16u);  // A-source lane, M-tile 1 (pos 16..31)

    for (int s = wave; s < B; s += wavesTotal) {
        const float* xs = x   + (size_t)s * 256;
        float*       os = out + (size_t)s * 256;

        // keep the 256 MB stream ahead of the wave
        {
            int sn = s + wavesTotal;
            if (sn < B)
                __builtin_prefetch((const char*)(x + (size_t)sn * 256) + lane * 32, 0, 1);
        }

        // ---- load: lane p holds all 8 channels of position p.
        //      Single-use stream -> non-temporal (TH=NT), coalesced b32s. ----
        float resid[8], h[8];
        #pragma unroll
        for (int c = 0; c < 8; ++c) resid[c] = __builtin_nontemporal_load(xs + c * 32 + lane);

        // ---- LN1 across positions, per channel (wave32 butterfly) ----
        #pragma unroll
        for (int c = 0; c < 8; ++c) {
            float v  = resid[c];
            float s1 = v, s2 = v * v;
            #pragma unroll
            for (int off = 16; off; off >>= 1) {
                s1 += __shfl_xor(s1, off, 32);
                s2 += __shfl_xor(s2, off, 32);
            }
            float mu  = s1 * 0.03125f;
            float var = s2 * 0.03125f - mu * mu;
            float inv = rsqrtf(var + LN_EPS);
            h[c] = (v - mu) * inv * g1 + be1;
        }

        // ---- matmul1 on the XDL pipe: (32 pos x 8 ch) @ W1^T + b1 ----
        v8f d0, d1;
        #pragma unroll
        for (int r = 0; r < 8; ++r) { d0[r] = cbias1; d1[r] = cbias1; }
        #pragma unroll
        for (int t = 0; t < 2; ++t) {
            const int src = t ? src1 : src0;
            #pragma unroll
            for (int k = 0; k < 2; ++k) {
                float t0 = __shfl(h[4 * k + 0], src, 32);
                float t1 = __shfl(h[4 * k + 1], src, 32);
                float t2 = __shfl(h[4 * k + 2], src, 32);
                float t3 = __shfl(h[4 * k + 3], src, 32);
                v2f a; a.x = hi ? t2 : t0; a.y = hi ? t3 : t1;   // A 16x4 layout
                v2f bb = k ? Bw1hi : Bw1lo;
                if (t == 0)
                    d0 = __builtin_amdgcn_wmma_f32_16x16x4_f32(false, a, false, bb,
                                                               (short)0, d0, false, false);
                else
                    d1 = __builtin_amdgcn_wmma_f32_16x16x4_f32(false, a, false, bb,
                                                               (short)0, d1, false, false);
            }
        }

        // ---- transpose D back to position-on-lane via per-wave LDS slab ----
        {
            const unsigned colbase = (hi ? 8u : 0u) * 16u + l15;
            #pragma unroll
            for (int r = 0; r < 8; ++r) tile[colbase + r * 16]       = d0[r];
            #pragma unroll
            for (int r = 0; r < 8; ++r) tile[colbase + 256 + r * 16] = d1[r];
        }
        asm volatile("s_wait_dscnt 0" ::: "memory");  // LDS in-order per wave + compiler barrier
        float hg[8];
        #pragma unroll
        for (int j = 0; j < 8; ++j) hg[j] = tile[lane * 16 + j];

        // ---- GELU, split u/v, LN2(v) ----
        float u[4], vln[4];
        #pragma unroll
        for (int j = 0; j < 4; ++j) u[j] = gelu_exact(hg[j]);
        #pragma unroll
        for (int i = 0; i < 4; ++i) {
            float v  = gelu_exact(hg[4 + i]);
            float s1 = v, s2 = v * v;
            #pragma unroll
            for (int off = 16; off; off >>= 1) {
                s1 += __shfl_xor(s1, off, 32);
                s2 += __shfl_xor(s2, off, 32);
            }
            float mu  = s1 * 0.03125f;
            float var = s2 * 0.03125f - mu * mu;
            float inv = rsqrtf(var + LN_EPS);
            vln[i] = (v - mu) * inv * g2 + be2;
        }

        // ---- conv1d (4->4, k=5, pad 2) along positions: lane shifts ----
        float cvo[4];
        #pragma unroll
        for (int o = 0; o < 4; ++o) cvo[o] = cb[o];
        #pragma unroll
        for (int i = 0; i < 4; ++i) {
            float c0 = vln[i];
            float m2 = __shfl(c0, (int)lane - 2, 32); m2 = (lane >= 2u)  ? m2 : 0.f;
            float m1 = __shfl(c0, (int)lane - 1, 32); m1 = (lane >= 1u)  ? m1 : 0.f;
            float p1 = __shfl(c0, (int)lane + 1, 32); p1 = (lane <= 30u) ? p1 : 0.f;
            float p2 = __shfl(c0, (int)lane + 2, 32); p2 = (lane <= 29u) ? p2 : 0.f;
            #pragma unroll
            for (int o = 0; o < 4; ++o) {
                cvo[o] = fmaf(cw[o][i][0], m2, cvo[o]);
                cvo[o] = fmaf(cw[o][i][1], m1, cvo[o]);
                cvo[o] = fmaf(cw[o][i][2], c0, cvo[o]);
                cvo[o] = fmaf(cw[o][i][3], p1, cvo[o]);
                cvo[o] = fmaf(cw[o][i][4], p2, cvo[o]);
            }
        }

        // ---- spatial gate ----
        float gg[4];
        #pragma unroll
        for (int i = 0; i < 4; ++i) gg[i] = u[i] * cvo[i];

        // ---- matmul2 on the XDL pipe: (32 pos x 4 ch) @ W2^T + b2 (K=4, one step) ----
        v8f e0, e1;
        #pragma unroll
        for (int r = 0; r < 8; ++r) { e0[r] = cbias2; e1[r] = cbias2; }
        #pragma unroll
        for (int t = 0; t < 2; ++t) {
            const int src = t ? src1 : src0;
            float t0 = __shfl(gg[0], src, 32);
            float t1 = __shfl(gg[1], src, 32);
            float t2 = __shfl(gg[2], src, 32);
            float t3 = __shfl(gg[3], src, 32);
            v2f a; a.x = hi ? t2 : t0; a.y = hi ? t3 : t1;
            if (t == 0)
                e0 = __builtin_amdgcn_wmma_f32_16x16x4_f32(false, a, false, Bw2,
                                                           (short)0, e0, false, false);
            else
                e1 = __builtin_amdgcn_wmma_f32_16x16x4_f32(false, a, false, Bw2,
                                                           (short)0, e1, false, false);
        }

        // ---- transpose back, GELU, residual, store (non-temporal, coalesced) ----
        {
            const unsigned colbase = (hi ? 8u : 0u) * 16u + l15;
            #pragma unroll
            for (int r = 0; r < 8; ++r) tile[colbase + r * 16]       = e0[r];
            #pragma unroll
            for (int r = 0; r < 8; ++r) tile[colbase + 256 + r * 16] = e1[r];
        }
        asm volatile("s_wait_dscnt 0" ::: "memory");
        #pragma unroll
        for (int c = 0; c < 8; ++c) {
            float val = tile[lane * 16 + c];
            __builtin_nontemporal_store(gelu_exact(val) + resid[c], os + c * 32 + lane);
        }
    }
}

extern "C" void kernel_launch(void* const* d_in, const int* in_sizes, int n_in,
                              void* d_out, int out_size, void* d_ws, size_t ws_size,
                              hipStream_t stream) {
    const float* x      = (const float*)d_in[0];
    const float* ln1_g  = (const float*)d_in[1];
    const float* ln1_b  = (const float*)d_in[2];
    const float* W1     = (const float*)d_in[3];
    const float* b1     = (const float*)d_in[4];
    const float* sgu_g  = (const float*)d_in[5];
    const float* sgu_b  = (const float*)d_in[6];
    const float* conv_w = (const float*)d_in[7];
    const float* conv_b = (const float*)d_in[8];
    const float* W2     = (const float*)d_in[9];
    const float* b2     = (const float*)d_in[10];
    float* out = (float*)d_out;

    const int B = in_sizes[0] / 256;            // samples of 8x32 f32
    const int threads = 256;                    // 8 wave32 per block
    int blocks = 1024;                          // persistent waves, grid-stride by wave
    const int wavesPerBlock = threads / 32;
    if (blocks * wavesPerBlock > B) blocks = (B + wavesPerBlock - 1) / wavesPerBlock;
    if (blocks < 1) blocks = 1;

    gmlp_fused_kernel<<<blocks, threads, 0, stream>>>(
        x, ln1_g, ln1_b, W1, b1, sgu_g, sgu_b, conv_w, conv_b, W2, b2, out, B);
}